// ChaosFEXLayer_7928509629117
// MI455X (gfx1250) — compile-verified
//
#include <hip/hip_runtime.h>
#include <math.h>

#define B_ROWS  128
#define D_COLS  2048
#define N_NEUR  512
#define T_ITER  1000
#define NBINS   20
#define HSTRIDE 21   // 20 bins + 1 pad dword -> odd stride, conflict-friendly

// ---- gfx1250 async global->LDS path (guarded; falls back to plain loads) ----
#if defined(__has_builtin)
#if __has_builtin(__builtin_amdgcn_global_load_async_to_lds_b128)
#define CHAOS_ASYNC 1
#endif
#endif
#ifndef CHAOS_ASYNC
#define CHAOS_ASYNC 0
#endif

#if CHAOS_ASYNC
typedef int chaos_v4i __attribute__((ext_vector_type(4)));
typedef chaos_v4i __attribute__((address_space(1)))* chaos_gp;  // global (AS1)
typedef chaos_v4i __attribute__((address_space(3)))* chaos_lp;  // LDS (AS3)

__device__ __forceinline__ void chaos_wait_async0() {
#if __has_builtin(__builtin_amdgcn_s_wait_asynccnt)
  __builtin_amdgcn_s_wait_asynccnt(0);
#else
  asm volatile("s_wait_asynccnt 0" ::: "memory");
#endif
}
#endif

// ---------------------------------------------------------------------------
// Kernel 1: per-row min/max of x (128 rows x 2048 cols). One block per row.
// ---------------------------------------------------------------------------
__global__ __launch_bounds__(256) void row_minmax_kernel(
    const float* __restrict__ x, float* __restrict__ mn, float* __restrict__ mx)
{
  __shared__ float smin[256];
  __shared__ float smax[256];
#if CHAOS_ASYNC
  __shared__ float row[D_COLS];
#endif
  const int tid = threadIdx.x;
  const int b   = blockIdx.x;
  const float* src = x + (size_t)b * D_COLS;

  float lmin =  3.402823466e38f;
  float lmax = -3.402823466e38f;

#if CHAOS_ASYNC
  // Stage the 8KB row into LDS with async b128 copies (2 iters x 256 lanes x 16B)
#pragma unroll
  for (int it = 0; it < D_COLS / (256 * 4); ++it) {
    const int off = (tid + it * 256) * 4;  // float index, 16B granularity
    __builtin_amdgcn_global_load_async_to_lds_b128(
        (chaos_gp)(src + off),
        (chaos_lp)(row + off),
        0, 0);
  }
  chaos_wait_async0();
  __syncthreads();
  for (int i = tid; i < D_COLS; i += 256) {
    float v = row[i];
    lmin = fminf(lmin, v);
    lmax = fmaxf(lmax, v);
  }
#else
  for (int i = tid * 4; i < D_COLS; i += 256 * 4) {
    float4 v = reinterpret_cast<const float4*>(src)[i >> 2];
    lmin = fminf(fminf(fminf(lmin, v.x), fminf(v.y, v.z)), v.w);
    lmax = fmaxf(fmaxf(fmaxf(lmax, v.x), fmaxf(v.y, v.z)), v.w);
  }
#endif

  smin[tid] = lmin;
  smax[tid] = lmax;
  __syncthreads();
  for (int sft = 128; sft > 0; sft >>= 1) {
    if (tid < sft) {
      smin[tid] = fminf(smin[tid], smin[tid + sft]);
      smax[tid] = fmaxf(smax[tid], smax[tid + sft]);
    }
    __syncthreads();
  }
  if (tid == 0) {
    mn[b] = smin[0];
    mx[b] = smax[0];
  }
}

// ---------------------------------------------------------------------------
// Kernel 2: chaos trajectory + features. One thread per (batch, neuron).
// Grid: (N_NEUR/256, B_ROWS), block 256.
// ---------------------------------------------------------------------------
__global__ __launch_bounds__(256) void chaos_features_kernel(
    const float* __restrict__ x, const float* __restrict__ rmn,
    const float* __restrict__ rmx, float* __restrict__ out)
{
  __shared__ unsigned int hist[256 * HSTRIDE];  // 21504 B

  const int tid = threadIdx.x;
  const int b   = blockIdx.y;
  const int n   = blockIdx.x * 256 + tid;

  unsigned int* h = &hist[tid * HSTRIDE];
#pragma unroll
  for (int k = 0; k < NBINS; ++k) h[k] = 0u;

  // np.linspace(0, 2047, 512) in float64, truncated toward zero; endpoint forced.
  const double step = 2047.0 / 511.0;
  const int idx = (n == N_NEUR - 1) ? (D_COLS - 1) : (int)((double)n * step);

  const float* xrow = x + (size_t)b * D_COLS;
  __builtin_prefetch(xrow + idx, 0, 0);  // global_prefetch_b8 on gfx1250

  const float mnv = rmn[b];
  const float mxv = rmx[b];
  const float v   = xrow[idx];
  float s = (v - mnv) / (mxv - mnv + 1e-10f);
  s = fminf(fmaxf(s, 0.01f), 0.99f);  // initial condition (bit-exact vs ref)

  // step 0 (ic) contributes to energy + histogram + threshold state
  float sumsq = __fmul_rn(s, s);
  {
    int b0 = (int)__fmul_rn(s, 20.0f);
    b0 = b0 < 0 ? 0 : (b0 > NBINS - 1 ? NBINS - 1 : b0);
    atomicAdd(&h[b0], 1u);  // ds_add_u32 (no return)
  }
  bool prev = s > 0.5f;

  int count = 0, first = 0, last = 0;

#pragma unroll 4
  for (int i = 0; i < T_ITER - 1; ++i) {
    // s_{n+1} = (s + 0.1*s*s) mod 1  -- no FMA contraction, mod = t - trunc(t)
    float m = __fmul_rn(0.1f, s);
    m = __fmul_rn(m, s);
    float t = __fadd_rn(s, m);
    s = t - truncf(t);

    const bool above = s > 0.5f;
    const bool cross = above && !prev;
    prev = above;

    first = (cross && (count == 0)) ? i : first;
    last  = cross ? i : last;
    count += cross ? 1 : 0;

    sumsq = __fadd_rn(sumsq, __fmul_rn(s, s));

    int bb = (int)__fmul_rn(s, 20.0f);
    bb = bb > NBINS - 1 ? NBINS - 1 : bb;
    atomicAdd(&h[bb], 1u);
  }

  // ---- features ----
  const float me  = sumsq / 1000.0f;
  const float mfr = (float)count / 1000.0f;
  const float mft = (count > 1)
                        ? ((float)(last - first) / (float)(count - 1))
                        : 1000.0f;

  // entropy of 20-bin density histogram:
  // hist = c / (T*(1/NBINS)) == c / 50.0f ; p = hist/(sum+1e-10);
  // q = p+1e-10; q /= sum(q); ent = -sum q*log q
  float hs[NBINS];
  float hsum = 0.0f;
#pragma unroll
  for (int k = 0; k < NBINS; ++k) {
    float hv = (float)h[k] / 50.0f;
    hs[k] = hv;
    hsum = __fadd_rn(hsum, hv);
  }
  float q[NBINS];
  float qs = 0.0f;
#pragma unroll
  for (int k = 0; k < NBINS; ++k) {
    float p = hs[k] / (hsum + 1e-10f);
    q[k] = p + 1e-10f;
    qs = __fadd_rn(qs, q[k]);
  }
  float ent = 0.0f;
#pragma unroll
  for (int k = 0; k < NBINS; ++k) {
    float r = q[k] / qs;
    ent = __fadd_rn(ent, -__fmul_rn(r, logf(r)));
  }

  // out[b, 4n .. 4n+3] = {mft, mfr, me, ment}; coalesced 16B store
  float4 f;
  f.x = mft; f.y = mfr; f.z = me; f.w = ent;
  reinterpret_cast<float4*>(out)[(size_t)b * N_NEUR + n] = f;
}

// ---------------------------------------------------------------------------
extern "C" void kernel_launch(void* const* d_in, const int* in_sizes, int n_in,
                              void* d_out, int out_size, void* d_ws, size_t ws_size,
                              hipStream_t stream) {
  (void)in_sizes; (void)n_in; (void)out_size; (void)ws_size;
  const float* x = (const float*)d_in[0];
  float* out = (float*)d_out;
  float* mn = (float*)d_ws;           // 128 floats
  float* mx = mn + B_ROWS;            // 128 floats

  hipLaunchKernelGGL(row_minmax_kernel, dim3(B_ROWS), dim3(256), 0, stream,
                     x, mn, mx);
  hipLaunchKernelGGL(chaos_features_kernel, dim3(N_NEUR / 256, B_ROWS),
                     dim3(256), 0, stream, x, mn, mx, out);
}